// MonarchAttention_56118042690156
// MI455X (gfx1250) — compile-verified
//
#include <hip/hip_runtime.h>

#define S    2048
#define DIM  768
#define H    12
#define D    64
#define F    32
#define EPSV 1e-6f
#define SCL  0.125f   // 64^-0.5

typedef float v2f __attribute__((ext_vector_type(2)));
typedef float v8f __attribute__((ext_vector_type(8)));
typedef unsigned int v4u __attribute__((ext_vector_type(4)));
typedef int v8i __attribute__((ext_vector_type(8)));
typedef int v4i __attribute__((ext_vector_type(4)));

static __device__ inline v8f wmma4(float a0, float a1, float b0, float b1, v8f c) {
  v2f a = {a0, a1};
  v2f b = {b0, b1};
  // V_WMMA_F32_16X16X4_F32: D = A(16x4) x B(4x16) + C(16x16), fp32 exact path
  return __builtin_amdgcn_wmma_f32_16x16x4_f32(false, a, false, b, (short)0, c,
                                               false, false);
}

static __device__ inline float wave_sum32(float v) {
  v += __shfl_xor(v, 16, 32);
  v += __shfl_xor(v, 8, 32);
  v += __shfl_xor(v, 4, 32);
  v += __shfl_xor(v, 2, 32);
  v += __shfl_xor(v, 1, 32);
  return v;
}

// TDM: async 2D tile load (64 rows x 64 f32, row stride 768 f32) -> packed LDS
static __device__ inline void tdm_load_v(const float* Vg, int h, int f,
                                         unsigned int lds_off) {
  unsigned long long ga =
      (unsigned long long)(uintptr_t)(Vg + (size_t)(f * 64) * DIM + h * D);
  // D# group0: count=1 | lds_addr | global_addr[56:0] | type=2
  v4u g0 = {1u, lds_off, (unsigned int)ga,
            (unsigned int)(ga >> 32) | 0x80000000u};
  // D# group1: data_size=4B; tensor_dim0=tensor_dim1=65536 (no OOB clamp);
  // tile_dim0=64, tile_dim1=64; tensor_dim0_stride=768
  v8i g1 = {0x00020000, 0, 1, (64 << 16) | 1, 64, 768, 0, 0};
  v4i gz4 = {0, 0, 0, 0};          // 2D tensor: groups 2/3 unused
  v8i gz8 = {0, 0, 0, 0, 0, 0, 0, 0};
  __builtin_amdgcn_tensor_load_to_lds(g0, g1, gz4, gz4, gz8, 0);
}

// -------- Kernel 1: Y = X @ W^T + b (2048x768 @ 768x768), 4 n-tiles/wave ----
__global__ __launch_bounds__(256) void k_proj(const float* __restrict__ X,
                                              const float* __restrict__ W,
                                              const float* __restrict__ Bv,
                                              float* __restrict__ OUT) {
  const int lid  = threadIdx.x & 31;
  const int wave = threadIdx.x >> 5;
  const int half = lid >> 4;
  const int ln   = lid & 15;
  const int m0   = (blockIdx.x * 8 + wave) * 16;
  const int n0   = blockIdx.y * 64;
  v8f acc0 = {0, 0, 0, 0, 0, 0, 0, 0};
  v8f acc1 = acc0, acc2 = acc0, acc3 = acc0;
  const float* ap = X + (size_t)(m0 + ln) * DIM + 2 * half;
  const float* bp = W + (size_t)(n0 + ln) * DIM + 2 * half;
#pragma unroll 2
  for (int kk = 0; kk < DIM; kk += 4) {
    float2 av = *(const float2*)(ap + kk);
    float2 b0 = *(const float2*)(bp + kk);
    float2 b1 = *(const float2*)(bp + 16 * DIM + kk);
    float2 b2 = *(const float2*)(bp + 32 * DIM + kk);
    float2 b3 = *(const float2*)(bp + 48 * DIM + kk);
    acc0 = wmma4(av.x, av.y, b0.x, b0.y, acc0);  // A fragment reused 4x
    acc1 = wmma4(av.x, av.y, b1.x, b1.y, acc1);
    acc2 = wmma4(av.x, av.y, b2.x, b2.y, acc2);
    acc3 = wmma4(av.x, av.y, b3.x, b3.y, acc3);
  }
  const float bb0 = Bv[n0 + ln];
  const float bb1 = Bv[n0 + 16 + ln];
  const float bb2 = Bv[n0 + 32 + ln];
  const float bb3 = Bv[n0 + 48 + ln];
#pragma unroll
  for (int r = 0; r < 8; ++r) {
    int m = m0 + r + 8 * half;
    float* o = OUT + (size_t)m * DIM + n0 + ln;
    o[0]  = acc0[r] + bb0;
    o[16] = acc1[r] + bb1;
    o[32] = acc2[r] + bb2;
    o[48] = acc3[r] + bb3;
  }
}

// ------------- Kernel 2: in-place RMSNorm (+RoPE for lq, rk) ----------------
__global__ __launch_bounds__(256) void k_norm(float* lq, float* lk, float* rq,
                                              float* rk,
                                              const float* __restrict__ g_lq,
                                              const float* __restrict__ g_lk,
                                              const float* __restrict__ g_rq,
                                              const float* __restrict__ g_rk,
                                              const float* __restrict__ cosb,
                                              const float* __restrict__ sinb) {
  const int unit = blockIdx.x * 8 + (threadIdx.x >> 5);
  const int lid  = threadIdx.x & 31;
  const int s = unit / H, h = unit % H;
  const int d0 = 2 * lid;
  const size_t base = (size_t)s * DIM + h * D;
  const float c0 = cosb[s * D + d0], c1 = cosb[s * D + d0 + 1];
  const float s0 = sinb[s * D + d0], s1 = sinb[s * D + d0 + 1];
  float* bufs[4] = {lq, lk, rq, rk};
  const float* gs[4] = {g_lq, g_lk, g_rq, g_rk};
  const int dorope[4] = {1, 0, 0, 1};
#pragma unroll
  for (int q = 0; q < 4; ++q) {
    float* p = bufs[q] + base;
    float x0 = p[d0], x1 = p[d0 + 1];
    float ss = wave_sum32(x0 * x0 + x1 * x1);
    float rms = rsqrtf(ss * (1.0f / 64.0f) + EPSV);
    float y0 = x0 * rms * gs[q][d0];
    float y1 = x1 * rms * gs[q][d0 + 1];
    if (dorope[q]) {
      float t0 = y0 * c0 - y1 * s0;
      float t1 = y1 * c1 + y0 * s1;
      y0 = t0;
      y1 = t1;
    }
    p[d0] = y0;
    p[d0 + 1] = y1;
  }
}

// --------- Kernel 3: per-head 64x64 projection (+bias, optional RoPE) -------
__global__ __launch_bounds__(128) void k_blin(const float* __restrict__ X,
                                              const float* __restrict__ W,
                                              const float* __restrict__ Bv,
                                              float* __restrict__ OUT,
                                              const float* __restrict__ cosb,
                                              const float* __restrict__ sinb,
                                              int doRope) {
  const int lid  = threadIdx.x & 31;
  const int wave = threadIdx.x >> 5;
  const int half = lid >> 4;
  const int ln   = lid & 15;
  const int m0   = blockIdx.x * 16;
  const int h    = blockIdx.y;
  const int n0   = wave * 16;
  v8f acc = {0, 0, 0, 0, 0, 0, 0, 0};
  const float* ap = X + (size_t)(m0 + ln) * DIM + h * D + 2 * half;
  const float* wp = W + (size_t)h * D * D + n0 + ln;
#pragma unroll
  for (int kk = 0; kk < D; kk += 4) {
    int ko = kk + 2 * half;
    float2 av = *(const float2*)(ap + kk);
    float b0 = wp[(size_t)ko * D];
    float b1 = wp[(size_t)(ko + 1) * D];
    acc = wmma4(av.x, av.y, b0, b1, acc);
  }
  const int n = n0 + ln;
  const float bb = Bv[h * D + n];
#pragma unroll
  for (int r = 0; r < 8; ++r) {
    int m = m0 + r + 8 * half;
    float val = acc[r] + bb;
    float partner = __shfl_xor(val, 1, 32);  // pair element n^1, same row
    if (doRope) {
      float cv = cosb[m * D + n], sv = sinb[m * D + n];
      val = (n & 1) ? (val * cv + partner * sv) : (val * cv - partner * sv);
    }
    OUT[(size_t)m * DIM + h * D + n] = val;
  }
}

// --------- Kernel 4/5: 8-wide local attention + RoPE (both branches) --------
__global__ __launch_bounds__(256) void k_local(const float* __restrict__ Q,
                                               const float* __restrict__ K,
                                               const float* __restrict__ V,
                                               float* __restrict__ OUT,
                                               const float* __restrict__ cosb,
                                               const float* __restrict__ sinb,
                                               int kstep, int qmul) {
  const int unit = blockIdx.x * 8 + (threadIdx.x >> 5);
  const int lid  = threadIdx.x & 31;
  const int f = unit / 96;
  const int rr = unit % 96;
  const int u = rr / 12;
  const int h = rr % 12;
  const int qpos = f * 64 + u * qmul;
  const int d0 = 2 * lid;
  const float* qp = Q + (size_t)qpos * DIM + h * D;
  const float q0 = qp[d0], q1 = qp[d0 + 1];
  float sc[8];
#pragma unroll
  for (int j = 0; j < 8; ++j) {
    const float* kp = K + (size_t)(qpos + j * kstep) * DIM + h * D;
    float p = wave_sum32(q0 * kp[d0] + q1 * kp[d0 + 1]);
    sc[j] = p * SCL;
  }
  float mx = sc[0];
#pragma unroll
  for (int j = 1; j < 8; ++j) mx = fmaxf(mx, sc[j]);
  float e[8], den = 0.0f;
#pragma unroll
  for (int j = 0; j < 8; ++j) {
    e[j] = __expf(sc[j] - mx);
    den += e[j];
  }
  const float inv = 1.0f / den;
  float a0 = 0.0f, a1 = 0.0f;
#pragma unroll
  for (int j = 0; j < 8; ++j) {
    const float* vp = V + (size_t)(qpos + j * kstep) * DIM + h * D;
    float w = e[j] * inv;
    a0 += w * vp[d0];
    a1 += w * vp[d0 + 1];
  }
  const float c0 = cosb[qpos * D + d0], c1 = cosb[qpos * D + d0 + 1];
  const float s0 = sinb[qpos * D + d0], s1 = sinb[qpos * D + d0 + 1];
  const float o0 = a0 * c0 - a1 * s0;
  const float o1 = a1 * c1 + a0 * s1;
  float* op = OUT + (size_t)(f * 8 + u) * DIM + h * D;
  op[d0] = o0;
  op[d0 + 1] = o1;
}

// --------- Kernel 6/7: per-head Gram GEMM C[h] = A_h (Mx64) @ B_h^T ---------
__global__ __launch_bounds__(128) void k_gram(const float* __restrict__ A,
                                              const float* __restrict__ B,
                                              float* __restrict__ C, int N) {
  const int lid  = threadIdx.x & 31;
  const int wave = threadIdx.x >> 5;
  const int half = lid >> 4;
  const int ln   = lid & 15;
  const int m0   = blockIdx.x * 16;
  const int n0   = (blockIdx.y * 4 + wave) * 16;
  const int h    = blockIdx.z;
  v8f acc = {0, 0, 0, 0, 0, 0, 0, 0};
  const float* ap = A + (size_t)(m0 + ln) * DIM + h * D + 2 * half;
  const float* bp = B + (size_t)(n0 + ln) * DIM + h * D + 2 * half;
#pragma unroll
  for (int kk = 0; kk < D; kk += 4) {
    float2 av = *(const float2*)(ap + kk);
    float2 bv = *(const float2*)(bp + kk);
    acc = wmma4(av.x, av.y, bv.x, bv.y, acc);
  }
  const size_t M = (size_t)gridDim.x * 16;
  const size_t cb = (size_t)h * M * (size_t)N;
#pragma unroll
  for (int r = 0; r < 8; ++r) {
    int m = m0 + r + 8 * half;
    C[cb + (size_t)m * N + n0 + ln] = acc[r];
  }
}

// ---- Kernel 8: fused structured score + softmax + P@V -----------------------
// (TDM double-buffered V tiles into LDS; WMMA fp32 for P@V)
__global__ __launch_bounds__(256) void k_final(const float* __restrict__ Lm,
                                               const float* __restrict__ Rm,
                                               const float* __restrict__ Vg,
                                               const float* __restrict__ Mask,
                                               float* __restrict__ OUT) {
  __shared__ float sP[64 * 65];     // padded stride: conflict-free A-frag reads
  __shared__ float sVd[2][64 * 64]; // double-buffered TDM destination
  __shared__ float sMax[64];
  __shared__ float sInv[64];
  const int a = blockIdx.x;
  const int h = blockIdx.y;
  const int t = threadIdx.x;

  // ---- phase 1: per-row max and sum of exp over all 2048 columns ----
  const int r0 = t >> 2;
  const int sub = t & 3;
  const int jrow = r0 & 7;
  const float* Lrow = Lm + ((size_t)h * 2048 + a * 64 + r0) * 256;
  const float* Rrow = Rm + ((size_t)h * 256 + a * 8 + jrow) * 2048;
  const float* Mrow = Mask + (size_t)(a * 64 + r0) * 2048;
  float mx = -3.0e38f;
  for (int c = sub; c < 2048; c += 4) {
    int f8k = ((c >> 6) << 3) | ((c >> 3) & 7);
    float sc = Lrow[f8k] * Rrow[c] * SCL + Mrow[c];
    mx = fmaxf(mx, sc);
  }
  mx = fmaxf(mx, __shfl_xor(mx, 1, 32));
  mx = fmaxf(mx, __shfl_xor(mx, 2, 32));
  if (sub == 0) sMax[r0] = mx;
  __syncthreads();
  const float rowm = sMax[r0];
  float den = 0.0f;
  for (int c = sub; c < 2048; c += 4) {
    int f8k = ((c >> 6) << 3) | ((c >> 3) & 7);
    float sc = Lrow[f8k] * Rrow[c] * SCL + Mrow[c];
    den += __expf(sc - rowm);
  }
  den += __shfl_xor(den, 1, 32);
  den += __shfl_xor(den, 2, 32);
  if (sub == 0) sInv[r0] = 1.0f / den;
  __syncthreads();

  // ---- phase 2: O(64x64) = sum_f Pf(64x64) @ Vf(64x64), WMMA fp32 ----
  const int lid  = t & 31;
  const int wave = t >> 5;
  const int half = lid >> 4;
  const int ln   = lid & 15;
  const int id0 = wave * 2, id1 = id0 + 1;
  const int tm0 = (id0 >> 2) * 16, tn0 = (id0 & 3) * 16;
  const int tm1 = (id1 >> 2) * 16, tn1 = (id1 & 3) * 16;
  v8f acc0 = {0, 0, 0, 0, 0, 0, 0, 0};
  v8f acc1 = {0, 0, 0, 0, 0, 0, 0, 0};

  if (wave == 0)  // prime the pipeline: V tile for f=0
    tdm_load_v(Vg, h, 0, (unsigned int)(uintptr_t)&sVd[0][0]);

  for (int f = 0; f < 32; ++f) {
    const int buf = f & 1;
    if (wave == 0) __builtin_amdgcn_s_wait_tensorcnt(0);  // tile f landed
    __syncthreads();  // publish sVd[buf]; prior compute on sVd[buf^1] done
    if (wave == 0 && f + 1 < 32)  // async-prefetch next V tile via TDM
      tdm_load_v(Vg, h, f + 1, (unsigned int)(uintptr_t)&sVd[buf ^ 1][0]);
    const float* sV = &sVd[buf][0];
    for (int idx = t; idx < 4096; idx += 256) {
      int rrow = idx >> 6, cl = idx & 63;
      float Lv = Lm[((size_t)h * 2048 + a * 64 + rrow) * 256 + f * 8 + (cl >> 3)];
      float Rv = Rm[((size_t)h * 256 + a * 8 + (rrow & 7)) * 2048 + f * 64 + cl];
      float mv = Mask[(size_t)(a * 64 + rrow) * 2048 + f * 64 + cl];
      float sc = Lv * Rv * SCL + mv;
      sP[rrow * 65 + cl] = __expf(sc - sMax[rrow]) * sInv[rrow];
    }
    __syncthreads();
#pragma unroll
    for (int kk = 0; kk < 64; kk += 4) {
      int ko = kk + 2 * half;
      float a0 = sP[(tm0 + ln) * 65 + ko];
      float a1 = sP[(tm0 + ln) * 65 + ko + 1];
      float b0 = sV[ko * 64 + tn0 + ln];
      float b1 = sV[(ko + 1) * 64 + tn0 + ln];
      acc0 = wmma4(a0, a1, b0, b1, acc0);
      a0 = sP[(tm1 + ln) * 65 + ko];
      a1 = sP[(tm1 + ln) * 65 + ko + 1];
      b0 = sV[ko * 64 + tn1 + ln];
      b1 = sV[(ko + 1) * 64 + tn1 + ln];
      acc1 = wmma4(a0, a1, b0, b1, acc1);
    }
  }
#pragma unroll
  for (int r = 0; r < 8; ++r) {
    int m0r = tm0 + r + 8 * half;
    int m1r = tm1 + r + 8 * half;
    OUT[(size_t)(a * 64 + m0r) * DIM + h * D + tn0 + ln] = acc0[r];
    OUT[(size_t)(a * 64 + m1r) * DIM + h * D + tn1 + ln] = acc1[r];
  }
}

// ---------------------------------------------------------------------------
extern "C" void kernel_launch(void* const* d_in, const int* in_sizes, int n_in,
                              void* d_out, int out_size, void* d_ws,
                              size_t ws_size, hipStream_t stream) {
  (void)in_sizes; (void)n_in; (void)out_size; (void)ws_size;
  const float* x    = (const float*)d_in[0];
  const float* cosb = (const float*)d_in[1];
  const float* sinb = (const float*)d_in[2];
  const float* mask = (const float*)d_in[3];
  const float* w_lq = (const float*)d_in[4];  const float* b_lq = (const float*)d_in[5];
  const float* w_lk = (const float*)d_in[6];  const float* b_lk = (const float*)d_in[7];
  const float* w_rq = (const float*)d_in[8];  const float* b_rq = (const float*)d_in[9];
  const float* w_rk = (const float*)d_in[10]; const float* b_rk = (const float*)d_in[11];
  const float* w_v  = (const float*)d_in[12]; const float* b_v  = (const float*)d_in[13];
  const float* w_lkq = (const float*)d_in[14]; const float* b_lkq = (const float*)d_in[15];
  const float* w_lkk = (const float*)d_in[16]; const float* b_lkk = (const float*)d_in[17];
  const float* w_lkv = (const float*)d_in[18]; const float* b_lkv = (const float*)d_in[19];
  const float* w_rqq = (const float*)d_in[20]; const float* b_rqq = (const float*)d_in[21];
  const float* w_rqk = (const float*)d_in[22]; const float* b_rqk = (const float*)d_in[23];
  const float* w_rqv = (const float*)d_in[24]; const float* b_rqv = (const float*)d_in[25];
  const float* g_lq = (const float*)d_in[26];
  const float* g_lk = (const float*)d_in[27];
  const float* g_rq = (const float*)d_in[28];
  const float* g_rk = (const float*)d_in[29];

  float* ws = (float*)d_ws;
  const size_t SZ = (size_t)S * DIM;
  float* lq  = ws + 0 * SZ;   // raw -> (norm+rope) in place
  float* lk  = ws + 1 * SZ;   // raw -> norm in place
  float* rq  = ws + 2 * SZ;   // raw -> norm in place
  float* rk  = ws + 3 * SZ;   // raw -> (norm+rope) in place
  float* v   = ws + 4 * SZ;
  float* lkq = ws + 5 * SZ;
  float* lkk = ws + 6 * SZ;
  float* lkv = ws + 7 * SZ;
  float* rqq = ws + 8 * SZ;
  float* rqk = ws + 9 * SZ;
  float* rqv = ws + 10 * SZ;
  float* lk_out = ws + 11 * SZ;            // (256, H, D)
  float* rq_out = lk_out + 256 * DIM;      // (256, H, D)
  float* Lmat = rq_out + 256 * DIM;        // (H, 2048, 256)
  float* Rmat = Lmat + (size_t)H * 2048 * 256;  // (H, 256, 2048)
  float* out = (float*)d_out;

  dim3 gp(S / 128, DIM / 64);  // 16 x 12; 8 waves/block, 4 n-tiles per wave
  k_proj<<<gp, 256, 0, stream>>>(x, w_lq, b_lq, lq);
  k_proj<<<gp, 256, 0, stream>>>(x, w_lk, b_lk, lk);
  k_proj<<<gp, 256, 0, stream>>>(x, w_rq, b_rq, rq);
  k_proj<<<gp, 256, 0, stream>>>(x, w_rk, b_rk, rk);
  k_proj<<<gp, 256, 0, stream>>>(x, w_v, b_v, v);

  k_norm<<<dim3(S * H / 8), 256, 0, stream>>>(lq, lk, rq, rk, g_lq, g_lk, g_rq,
                                              g_rk, cosb, sinb);

  dim3 gb(S / 16, H);  // 128 x 12, 128 thr (4 waves over 4 n-tiles)
  k_blin<<<gb, 128, 0, stream>>>(lk, w_lkq, b_lkq, lkq, cosb, sinb, 1);
  k_blin<<<gb, 128, 0, stream>>>(lk, w_lkk, b_lkk, lkk, cosb, sinb, 1);
  k_blin<<<gb, 128, 0, stream>>>(lk, w_lkv, b_lkv, lkv, cosb, sinb, 0);
  k_blin<<<gb, 128, 0, stream>>>(rq, w_rqq, b_rqq, rqq, cosb, sinb, 1);
  k_blin<<<gb, 128, 0, stream>>>(rq, w_rqk, b_rqk, rqk, cosb, sinb, 1);
  k_blin<<<gb, 128, 0, stream>>>(rq, w_rqv, b_rqv, rqv, cosb, sinb, 0);

  // L branch: q at f*64+i*8 (j=0), keys stride 1; R branch: q at f*64+j (i=0),
  // keys stride 8. 3072 (f,u,h) wave-units each.
  k_local<<<dim3(384), 256, 0, stream>>>(lkq, lkk, lkv, lk_out, cosb, sinb, 1, 8);
  k_local<<<dim3(384), 256, 0, stream>>>(rqq, rqk, rqv, rq_out, cosb, sinb, 8, 1);

  // Lmat[h][s][f*8+k] = lq . lk_out ; Rmat[h][a*8+j][col] = rq_out . rk
  k_gram<<<dim3(128, 4, H), 128, 0, stream>>>(lq, lk_out, Lmat, 256);
  k_gram<<<dim3(16, 32, H), 128, 0, stream>>>(rq_out, rk, Rmat, 2048);

  k_final<<<dim3(F, H), 256, 0, stream>>>(Lmat, Rmat, v, mask, out);
}